// TokenCombiner_70523363000736
// MI455X (gfx1250) — compile-verified
//
#include <hip/hip_runtime.h>

// Token-combine permutation copy for MI455X (gfx1250).
// Pure data movement: 512 MiB of HBM traffic, 0 FLOPs -> memory-roofline
// (~22.5 us at 23.3 TB/s). One block per output row (4096 fp32 = 16 KiB).
//
// Primary path (gfx1250 async data movers): per-lane
//   GLOBAL_LOAD_ASYNC_TO_LDS_B128  (HBM -> LDS, no VGPR payload, ASYNCcnt)
//   s_wait_asynccnt 0
//   GLOBAL_STORE_ASYNC_FROM_LDS_B128 (LDS -> HBM)
// Each lane owns disjoint LDS ranges -> no barrier needed; s_endpgm's
// implicit wait-idle drains the final async stores.
//
// Fallback path: register-staged global_load_b128/global_store_b128 with
// non-temporal hints (streaming data, keep the 192 MB L2 clean).

typedef __attribute__((ext_vector_type(4))) float f32x4;
typedef __attribute__((ext_vector_type(4))) int   i32x4;

#define HIDDEN        4096                  // fixed by the reference
#define TPB           256                   // 8 wave32 per block
#define V_PER_ROW     (HIDDEN / 4)          // 1024 float4 per row
#define V_PER_THREAD  (V_PER_ROW / TPB)     // 4 float4 per thread

#if defined(__has_builtin)
#if __has_builtin(__builtin_amdgcn_global_load_async_to_lds_b128) && \
    __has_builtin(__builtin_amdgcn_global_store_async_from_lds_b128) && \
    __has_builtin(__builtin_amdgcn_s_wait_asynccnt)
#define USE_ASYNC_LDS 1
#endif
#endif

#if defined(USE_ASYNC_LDS)
// Builtin prototype (from hipcc diagnostics): global arg is
// 'int __vector(4) addrspace(1)*', LDS arg is the addrspace(3) counterpart.
typedef __attribute__((address_space(1))) i32x4 gv4i;   // global int4
typedef __attribute__((address_space(3))) i32x4 lv4i;   // LDS int4
#endif

// NS > 0: compile-time split count (fully unrolled scalar lookup).
// NS == 0: runtime split count (generic fallback).
template <int NS>
__global__ __launch_bounds__(TPB) void token_combine_kernel(
    const float* __restrict__ inp,        // [in_rows, HIDDEN]
    const float* __restrict__ out_in,     // [out_rows, HIDDEN] fallback values
    const int*   __restrict__ in_so,      // [2, nsplits]: splits then offsets
    const int*   __restrict__ out_so,     // [2, nsplits]: splits then offsets
    float*       __restrict__ out,        // [out_rows, HIDDEN]
    int nsplits, int in_rows, int out_rows)
{
    const int row = (int)blockIdx.x;
    if (row >= out_rows) return;

    const int n = (NS > 0) ? NS : nsplits;

    // ---- uniform per-block chunk lookup (scalarizes to s_cmp/s_cselect) ----
    // j = last k with row >= out_off[k]  (== searchsorted(right)-1, clipped)
    int j = 0;
    if (NS > 0) {
#pragma unroll
        for (int k = 1; k < (NS > 0 ? NS : 1); ++k) {
            if (row >= out_so[NS + k]) j = k;
        }
    } else {
        for (int k = 1; k < nsplits; ++k) {
            if (row >= out_so[nsplits + k]) j = k;
        }
    }
    const int  pos   = row - out_so[n + j];   // may be negative (matches ref)
    const bool valid = pos < out_so[j];       // signed compare, like jnp

    const f32x4* __restrict__ src;
    if (valid) {
        long s = (long)in_so[n + j] + (long)pos;
        if (s < 0) s = 0;
        if (s > (long)in_rows - 1) s = (long)in_rows - 1;   // jnp.clip on src row
        src = (const f32x4*)(inp + s * (long)HIDDEN);
    } else {
        // padding row: keep the `out` buffer's value
        src = (const f32x4*)(out_in + (long)row * HIDDEN);
    }
    f32x4* __restrict__ dst = (f32x4*)(out + (long)row * HIDDEN);

    const int t = (int)threadIdx.x;

#if defined(USE_ASYNC_LDS)
    // ---- gfx1250 async data-mover path: HBM -> LDS -> HBM, no VGPR data ----
    __shared__ f32x4 stage[V_PER_ROW];      // 16 KiB per block
#pragma unroll
    for (int i = 0; i < V_PER_THREAD; ++i) {
        const int idx = t + i * TPB;
        __builtin_amdgcn_global_load_async_to_lds_b128(
            (gv4i*)(src + idx), (lv4i*)&stage[idx], 0, 0);
    }
    // Each lane reads back only the LDS bytes it wrote -> a same-wave
    // ASYNCcnt drain is the only ordering required (no barrier).
    __builtin_amdgcn_s_wait_asynccnt(0);
#pragma unroll
    for (int i = 0; i < V_PER_THREAD; ++i) {
        const int idx = t + i * TPB;
        __builtin_amdgcn_global_store_async_from_lds_b128(
            (gv4i*)(dst + idx), (lv4i*)&stage[idx], 0, 0);
    }
    // s_endpgm performs an implicit wait-idle -> async stores complete.
#else
    // ---- fallback: register-staged streaming copy with NT hints ----
#pragma unroll
    for (int i = 0; i < V_PER_THREAD; ++i) {
        f32x4 v = __builtin_nontemporal_load(src + t + i * TPB);
        __builtin_nontemporal_store(v, dst + t + i * TPB);
    }
#endif
}

extern "C" void kernel_launch(void* const* d_in, const int* in_sizes, int n_in,
                              void* d_out, int out_size, void* d_ws, size_t ws_size,
                              hipStream_t stream) {
    const float* inp    = (const float*)d_in[0];
    const float* out_in = (const float*)d_in[1];
    const int*   in_so  = (const int*)d_in[2];   // [2, nsplits] int32 on device
    const int*   out_so = (const int*)d_in[3];

    float* out = (float*)d_out;

    const int nsplits  = in_sizes[2] / 2;        // 8
    const int in_rows  = in_sizes[0] / HIDDEN;   // 16384
    const int out_rows = out_size    / HIDDEN;   // 16384

    dim3 grid((unsigned)out_rows), block(TPB);
    if (nsplits == 8) {
        token_combine_kernel<8><<<grid, block, 0, stream>>>(
            inp, out_in, in_so, out_so, out, nsplits, in_rows, out_rows);
    } else {
        token_combine_kernel<0><<<grid, block, 0, stream>>>(
            inp, out_in, in_so, out_so, out, nsplits, in_rows, out_rows);
    }
}